// Compressor_13460427505902
// MI455X (gfx1250) — compile-verified
//
#include <hip/hip_runtime.h>

// CDNA5 / gfx1250, wave32.
// 4096x4096 f32 -> 8x8 blockwise orthonormal DCT-II, per-block max-abs
// ("biggest", 512x512 f32) + int8 quantization (512x512x64, stored as float).
// d_out layout (float*): [0 .. 512*512) biggest, then 512*512*64 indices.
//
// One wave per 16x16 tile (= 4 DCT blocks): with Dd = diag(D,D) block-diagonal,
// Dd^T * T * Dd computes all four 8x8 DCTs at once. Both GEMMs are chains of
// V_WMMA_F32_16X16X4_F32 (full f32 precision). Roofline: ~132 MiB traffic at
// 23.3 TB/s ~ 5.7 us; FLOPs negligible -> memory bound, so f32 WMMA is free.

typedef __attribute__((ext_vector_type(2))) float v2f;
typedef __attribute__((ext_vector_type(8))) float v8f;

#define NB 512            // 4096/8 DCT blocks per dimension
#define TPR 256           // 16x16 tiles per row (4096/16)
#define WAVES_PER_BLOCK 8

// Orthonormal 8-pt DCT-II constants: A0 = sqrt(1/8), Bk = 0.5*cos(k*pi/16)
#define A0 0.35355339059327373f
#define B1 0.49039264020161522f
#define B2 0.46193976625564337f
#define B3 0.41573480615127262f
#define B5 0.27778511650980114f
#define B6 0.19134171618254492f
#define B7 0.09754516100806417f

// DCT_PAIR[j][h] = { D[2j][h], D[2j+1][h] } with D[e][f] = scale(f)*cos(pi*(2e+1)*f/16)
static __device__ const v2f DCT_PAIR[4][8] = {
    {{A0, A0}, {B1, B3}, {B2, B6}, {B3, -B7}, {A0, -A0}, {B5, -B1}, {B6, -B2}, {B7, -B5}},
    {{A0, A0}, {B5, B7}, {-B6, -B2}, {-B1, -B5}, {-A0, A0}, {B7, B3}, {B2, -B6}, {B3, -B1}},
    {{A0, A0}, {-B7, -B5}, {-B2, -B6}, {B5, B1}, {A0, -A0}, {-B3, -B7}, {-B6, B2}, {B1, -B3}},
    {{A0, A0}, {-B3, -B1}, {B6, B2}, {B7, -B3}, {-A0, A0}, {B1, -B5}, {-B2, B6}, {B5, -B7}},
};

// Lane L <-> lane L^16 exchange: ds_swizzle group-of-32 SWAPX16
// (offset[14:10]=xor=0x10, or=0, and=0x1f -> 0x401F). No LDS memory touched.
__device__ __forceinline__ float swap16(float x) {
  return __int_as_float(__builtin_amdgcn_ds_swizzle(__float_as_int(x), 0x401f));
}

// DPP8 xor-lane selects (lane i reads lane i^k within its 8-lane group)
#define DPP8_XOR1 0x00DE54C1  // 1,0,3,2,5,4,7,6
#define DPP8_XOR2 0x00B3E21A  // 2,3,0,1,6,7,4,5
#define DPP8_XOR4 0x00688FAC  // 4,5,6,7,0,1,2,3
#define DPP8F(x, sel) __int_as_float(__builtin_amdgcn_mov_dpp8(__float_as_int(x), (sel)))

__global__ __launch_bounds__(256) void dct8_quant_kernel(
    const float* __restrict__ x, float* __restrict__ biggest,
    float* __restrict__ indices) {
  const int lane = threadIdx.x & 31;
  const int wave = threadIdx.x >> 5;
  const int tile = blockIdx.x * WAVES_PER_BLOCK + wave;
  const int ti = tile / TPR;
  const int tj = tile - ti * TPR;
  const int half = lane >> 4;  // 0: lanes 0-15, 1: lanes 16-31
  const int n = lane & 15;     // column (B/C/D) or row (A) index
  const int h = lane & 7;      // frequency column within an 8x8 block
  const bool lo8 = (n < 8);

  // Per-lane Dd chunks (mirror A/B layout symmetry: same values serve as the
  // B-operand of Dd in stage 1 and the A-operand of Dd^T in stage 2).
  // dd[k] = (Dd[4k+2*half, n], Dd[4k+2*half+1, n]); block-diagonal zeros
  // reduce to two table pairs masked by n<8 vs n>=8.
  const v2f p0 = DCT_PAIR[half][h];      // rows 2*half, 2*half+1 of D
  const v2f p1 = DCT_PAIR[2 + half][h];  // rows 4+2*half, 5+2*half of D
  const v2f zz = {0.0f, 0.0f};
  v2f dd[4];
  dd[0] = lo8 ? p0 : zz;
  dd[1] = lo8 ? p1 : zz;
  dd[2] = lo8 ? zz : p0;
  dd[3] = lo8 ? zz : p1;

  // A-operand chunks of the 16x16 tile T: lane holds row n; chunk k is the
  // aligned 8-byte pair at columns 4k+2*half.
  const float* rowp = x + (size_t)(ti * 16 + n) * 4096 + (size_t)tj * 16;
  v2f a1[4];
#pragma unroll
  for (int k = 0; k < 4; ++k) a1[k] = *(const v2f*)(rowp + 4 * k + 2 * half);

  // Stage 1: U = T x Dd (4 chained K=4 WMMAs)
  v8f U = {};
#pragma unroll
  for (int k = 0; k < 4; ++k)
    U = __builtin_amdgcn_wmma_f32_16x16x4_f32(false, a1[k], false, dd[k],
                                              (short)0, U, false, false);

  // Re-shape U (C/D layout: lane = col N, VGPR r = row M = r + 8*half) into
  // B-operand chunks. Each half needs only 4 rows from the other half, so
  // pre-select what each side SENDS and use 4 exchanges total:
  //   lanes<16 receive U[8..9,12..13]; lanes>=16 receive U[2..3,6..7].
  const bool lo = (half == 0);
  const float w0 = swap16(lo ? (float)U[2] : (float)U[0]);  // lo: S0, hi: S2
  const float w1 = swap16(lo ? (float)U[3] : (float)U[1]);  // lo: S1, hi: S3
  const float w2 = swap16(lo ? (float)U[6] : (float)U[4]);  // lo: S4, hi: S6
  const float w3 = swap16(lo ? (float)U[7] : (float)U[5]);  // lo: S5, hi: S7
  v2f b2[4];
  b2[0].x = lo ? (float)U[0] : w0;  b2[0].y = lo ? (float)U[1] : w1;
  b2[1].x = lo ? (float)U[4] : w2;  b2[1].y = lo ? (float)U[5] : w3;
  b2[2].x = lo ? w0 : (float)U[2];  b2[2].y = lo ? w1 : (float)U[3];
  b2[3].x = lo ? w2 : (float)U[6];  b2[3].y = lo ? w3 : (float)U[7];

  // Stage 2: C = Dd^T x U (4 more K=4 WMMAs)
  v8f C = {};
#pragma unroll
  for (int k = 0; k < 4; ++k)
    C = __builtin_amdgcn_wmma_f32_16x16x4_f32(false, dd[k], false, b2[k],
                                              (short)0, C, false, false);

  // Each lane's 8 C values lie in exactly one 8x8 block (fixed N=n, M spans
  // one half). In-lane max-abs, then DPP8 xor-reduce across the 8 lanes
  // sharing the block (contiguous groups of 8 lanes).
  float m = 0.0f;
#pragma unroll
  for (int r = 0; r < 8; ++r) m = fmaxf(m, fabsf((float)C[r]));
  m = fmaxf(m, DPP8F(m, DPP8_XOR1));
  m = fmaxf(m, DPP8F(m, DPP8_XOR2));
  m = fmaxf(m, DPP8F(m, DPP8_XOR4));
  const float safe = (m == 0.0f) ? 1.0f : m;
  const float qscale = 127.0f / safe;

  const int bi = 2 * ti + half;            // DCT block row
  const int bj = 2 * tj + ((n >> 3) & 1);  // DCT block col
  if (h == 0) biggest[bi * NB + bj] = m;

  // indices[(bi,bj), g*8+h], g = r. rintf == jnp.round (RTNE); |q| <= 127 so
  // the reference's int8 cast never wraps.
  float* dst = indices + (((size_t)(bi * NB + bj)) << 6) + h;
#pragma unroll
  for (int r = 0; r < 8; ++r) dst[r * 8] = rintf((float)C[r] * qscale);
}

extern "C" void kernel_launch(void* const* d_in, const int* in_sizes, int n_in,
                              void* d_out, int out_size, void* d_ws,
                              size_t ws_size, hipStream_t stream) {
  (void)in_sizes; (void)n_in; (void)d_ws; (void)ws_size; (void)out_size;
  const float* x = (const float*)d_in[0];
  float* out = (float*)d_out;
  float* biggest = out;            // 512*512 floats
  float* indices = out + NB * NB;  // 512*512*64 floats
  // 256x256 tiles, 1 wave per tile, 8 waves per 256-thread block
  dct8_quant_kernel<<<(TPR * TPR) / WAVES_PER_BLOCK, 256, 0, stream>>>(
      x, biggest, indices);
}